// GCN_29703993819226
// MI455X (gfx1250) — compile-verified
//
#include <hip/hip_runtime.h>
#include <math.h>

// -------- CDNA5 WMMA fragment types (wave32) --------
typedef float v2f __attribute__((ext_vector_type(2)));  // A/B frag: 16x4 / 4x16 f32 = 2 VGPRs
typedef float v8f __attribute__((ext_vector_type(8)));  // C/D frag: 16x16 f32 = 8 VGPRs

// ---------------------------------------------------------------------------
// Degree / normalization kernels
// ---------------------------------------------------------------------------
__global__ __launch_bounds__(256) void k_fill_deg(float* __restrict__ deg, int n) {
  int i = blockIdx.x * blockDim.x + threadIdx.x;
  if (i < n) deg[i] = 1.0f;                       // self-loop contributes 1
}

__global__ __launch_bounds__(256) void k_accum_deg(const int* __restrict__ row,
                                                   float* __restrict__ deg, int e) {
  int i = blockIdx.x * blockDim.x + threadIdx.x;
  if (i < e) atomicAdd(&deg[row[i]], 1.0f);
}

__global__ __launch_bounds__(256) void k_rsqrt_inplace(float* __restrict__ deg, int n) {
  int i = blockIdx.x * blockDim.x + threadIdx.x;
  if (i < n) deg[i] = rsqrtf(deg[i]);             // deg >= 1 always (self loops)
}

// ---------------------------------------------------------------------------
// H = relu?(A) @ W   A:[M,K] row-major, W:[K,N] row-major, H:[M,N]
// One wave per 16x16 tile, V_WMMA_F32_16X16X4_F32, K stepped by 4.
// Requires M%16==0, N%16==0, K%4==0 (holds: M=100000, K=128, N in {128,16}).
// RELU is a compile-time flag so the hot loop is just loads + WMMA.
// ---------------------------------------------------------------------------
template <int RELU>
__global__ __launch_bounds__(256) void k_gemm_wmma_f32(
    const float* __restrict__ A, const float* __restrict__ W, float* __restrict__ H,
    int M, int N, int K, int nTiles) {
  const int waveInBlock = threadIdx.x >> 5;
  const int lane        = threadIdx.x & 31;
  const int tile        = blockIdx.x * 8 + waveInBlock;
  if (tile >= nTiles) return;                     // wave-uniform: EXEC stays all-ones

  const int ntilesN = N >> 4;
  const int m0 = (tile / ntilesN) << 4;
  const int n0 = (tile % ntilesN) << 4;
  const int half = lane >> 4;                     // 0: lanes 0-15, 1: lanes 16-31
  const int lr   = lane & 15;

  // A frag: lanes 0-15 hold K={k,k+1} of row m0+lr; lanes 16-31 hold K={k+2,k+3}
  const float* __restrict__ ap = A + (size_t)(m0 + lr) * K + 2 * half;
  // B frag mirrors A along K: b.x = W[kb][n0+lr], b.y = W[kb+1][n0+lr]
  const float* __restrict__ wp = W + (size_t)(2 * half) * N + n0 + lr;
  const size_t wstep = (size_t)4 * N;

  v8f c = {};
  #pragma unroll 8
  for (int k = 0; k < K; k += 4) {
    v2f a = *(const v2f*)(ap);                    // 8B-aligned pair load
    if (RELU) { a.x = fmaxf(a.x, 0.0f); a.y = fmaxf(a.y, 0.0f); }
    v2f b;
    b.x = wp[0];
    b.y = wp[N];
    ap += 4;
    wp += wstep;
    // 8 args: (neg_a, A, neg_b, B, c_mod, C, reuse_a, reuse_b)
    c = __builtin_amdgcn_wmma_f32_16x16x4_f32(false, a, false, b, (short)0, c,
                                              false, false);
  }
  // D layout: VGPR r -> row m0 + r + 8*half, col n0 + lr
  float* __restrict__ hrow = H + (size_t)(m0 + 8 * half) * N + n0 + lr;
  #pragma unroll
  for (int r = 0; r < 8; ++r) hrow[(size_t)r * N] = c[r];
}

// ---------------------------------------------------------------------------
// out[i,:] = bias + dinv[i]^2 * h[i,:]   (bias + self-loop term, no atomics)
// One thread per (node, 4-feature chunk); F = 1<<(shift+2).
// ---------------------------------------------------------------------------
__global__ __launch_bounds__(256) void k_init_out(
    const float* __restrict__ h, const float* __restrict__ dinv,
    const float* __restrict__ bias, float* __restrict__ out, int n, int F, int shift) {
  const int t    = blockIdx.x * blockDim.x + threadIdx.x;
  const int node = t >> shift;
  const int q    = t & ((1 << shift) - 1);
  if (node >= n) return;
  float s = dinv[node];
  s *= s;
  const float4 hv = ((const float4*)(h + (size_t)node * F))[q];
  const float4 bv = ((const float4*)bias)[q];
  float4 o = make_float4(bv.x + s * hv.x, bv.y + s * hv.y,
                         bv.z + s * hv.z, bv.w + s * hv.w);
  ((float4*)(out + (size_t)node * F))[q] = o;
}

// ---------------------------------------------------------------------------
// Edge scatter: out[row[e],:] += dinv[row]*dinv[col] * h[col[e],:]
// One thread per (edge, 4-feature chunk). L2-resident h/out -> atomics at L2.
// ---------------------------------------------------------------------------
__global__ __launch_bounds__(256) void k_scatter_edges(
    const int* __restrict__ row, const int* __restrict__ col,
    const float* __restrict__ dinv, const float* __restrict__ h,
    float* __restrict__ out, int e, int F, int shift) {
  const int t  = blockIdx.x * blockDim.x + threadIdx.x;
  const int ed = t >> shift;
  const int q  = t & ((1 << shift) - 1);
  if (ed >= e) return;
  const int r  = row[ed];
  const int cn = col[ed];
  const float nrm = dinv[r] * dinv[cn];
  const float4 hv = ((const float4*)(h + (size_t)cn * F))[q];
  float* __restrict__ op = out + (size_t)r * F + (q << 2);
  atomicAdd(op + 0, nrm * hv.x);
  atomicAdd(op + 1, nrm * hv.y);
  atomicAdd(op + 2, nrm * hv.z);
  atomicAdd(op + 3, nrm * hv.w);
}

// ---------------------------------------------------------------------------
// Row-wise log_softmax over 16 features.
// ---------------------------------------------------------------------------
__global__ __launch_bounds__(256) void k_log_softmax16(
    const float* __restrict__ in, float* __restrict__ out, int n) {
  const int i = blockIdx.x * blockDim.x + threadIdx.x;
  if (i >= n) return;
  const float* __restrict__ p = in + (size_t)i * 16;
  float v[16];
  float m = -INFINITY;
  #pragma unroll
  for (int j = 0; j < 16; ++j) { v[j] = p[j]; m = fmaxf(m, v[j]); }
  float s = 0.0f;
  #pragma unroll
  for (int j = 0; j < 16; ++j) s += expf(v[j] - m);
  const float ls = logf(s) + m;
  float* __restrict__ o = out + (size_t)i * 16;
  #pragma unroll
  for (int j = 0; j < 16; ++j) o[j] = v[j] - ls;
}

// ---------------------------------------------------------------------------
static inline int cdiv(long long a, int b) { return (int)((a + b - 1) / b); }

extern "C" void kernel_launch(void* const* d_in, const int* in_sizes, int n_in,
                              void* d_out, int out_size, void* d_ws, size_t ws_size,
                              hipStream_t stream) {
  const float* x  = (const float*)d_in[0];
  const int*   ei = (const int*)d_in[1];   // edge_index materializes as int32 (x64 off)
  const float* W1 = (const float*)d_in[2];
  const float* b1 = (const float*)d_in[3];
  const float* W2 = (const float*)d_in[4];
  const float* b2 = (const float*)d_in[5];
  const float* W3 = (const float*)d_in[6];
  const float* b3 = (const float*)d_in[7];

  const int DIN = 128, DHID = 128, DOUT = 16;
  const int N = in_sizes[0] / DIN;   // 100000
  const int E = in_sizes[1] / 2;     // 1600000
  const int* row = ei;               // edge_index[0]
  const int* col = ei + E;           // edge_index[1]

  // Workspace layout (floats): dinv[N] | H[N*128] | G[N*128]  (~103 MB total)
  float* dinv = (float*)d_ws;
  float* H    = dinv + N;                       // offset 400000 B, 16B aligned
  float* G    = H + (size_t)N * DHID;

  const int TB = 256;

  // --- symmetric normalization ---
  k_fill_deg<<<cdiv(N, TB), TB, 0, stream>>>(dinv, N);
  k_accum_deg<<<cdiv(E, TB), TB, 0, stream>>>(row, dinv, E);
  k_rsqrt_inplace<<<cdiv(N, TB), TB, 0, stream>>>(dinv, N);

  const int mTiles      = N / 16;               // 6250 (exact)
  const int tilesWide   = mTiles * (DHID / 16); // 50000 tiles, 8 waves/block
  const int tilesNarrow = mTiles * (DOUT / 16); // 6250 tiles

  // --- layer 1: h = x@W1 ; agg = b1 + selfloop + edges ---
  k_gemm_wmma_f32<0><<<cdiv(tilesWide, 8), TB, 0, stream>>>(x, W1, H, N, DHID, DIN, tilesWide);
  k_init_out<<<cdiv((long long)N * (DHID / 4), TB), TB, 0, stream>>>(H, dinv, b1, G, N, DHID, 5);
  k_scatter_edges<<<cdiv((long long)E * (DHID / 4), TB), TB, 0, stream>>>(row, col, dinv, H, G, E, DHID, 5);

  // --- layer 2: h = relu(G)@W2 (relu fused into A load) ---
  k_gemm_wmma_f32<1><<<cdiv(tilesWide, 8), TB, 0, stream>>>(G, W2, H, N, DHID, DHID, tilesWide);
  k_init_out<<<cdiv((long long)N * (DHID / 4), TB), TB, 0, stream>>>(H, dinv, b2, G, N, DHID, 5);
  k_scatter_edges<<<cdiv((long long)E * (DHID / 4), TB), TB, 0, stream>>>(row, col, dinv, H, G, E, DHID, 5);

  // --- layer 3: h = relu(G)@W3 (16-wide), then log_softmax ---
  k_gemm_wmma_f32<1><<<cdiv(tilesNarrow, 8), TB, 0, stream>>>(G, W3, H, N, DOUT, DHID, tilesNarrow);
  k_init_out<<<cdiv((long long)N * (DOUT / 4), TB), TB, 0, stream>>>(H, dinv, b3, G, N, DOUT, 2);
  k_scatter_edges<<<cdiv((long long)E * (DOUT / 4), TB), TB, 0, stream>>>(row, col, dinv, H, G, E, DOUT, 2);

  k_log_softmax16<<<cdiv(N, TB), TB, 0, stream>>>(G, (float*)d_out, N);
}